// SpringOptModule_44186623541278
// MI455X (gfx1250) — compile-verified
//
#include <hip/hip_runtime.h>

#define M_ALPHA 0.1f
#define EPS 1e-8f
#define TPB 256

#if __has_builtin(__builtin_amdgcn_global_load_async_to_lds_b64) && __has_builtin(__builtin_amdgcn_s_wait_asynccnt)
#define HAVE_ASYNC_LDS 1
typedef int v2i_t __attribute__((vector_size(8)));
typedef __attribute__((address_space(1))) v2i_t* g_v2i_ptr;
typedef __attribute__((address_space(3))) v2i_t* l_v2i_ptr;
#else
#define HAVE_ASYNC_LDS 0
#endif

__device__ __forceinline__ void atomAdd(float* a, float v) {
  __hip_atomic_fetch_add(a, v, __ATOMIC_RELAXED, __HIP_MEMORY_SCOPE_AGENT);
}

// wave32 + cross-wave block reduction; valid on thread 0
__device__ __forceinline__ float blockReduceSum(float v) {
#pragma unroll
  for (int off = 16; off > 0; off >>= 1) v += __shfl_down(v, off, 32);
  __shared__ float smem[TPB / 32];
  const int lane = threadIdx.x & 31, wid = threadIdx.x >> 5;
  if (lane == 0) smem[wid] = v;
  __syncthreads();
  if (wid == 0) {
    v = (lane < (TPB >> 5)) ? smem[lane] : 0.0f;
#pragma unroll
    for (int off = 4; off > 0; off >>= 1) v += __shfl_down(v, off, 32);
  }
  return v;
}

__device__ __forceinline__ float3 ld3(const float* __restrict__ base, int idx) {
  return ((const float3*)base)[idx];  // b96 gather
}

// ---------------- one-time setup kernels -----------------------------------

// md[i] = M_ALPHA * m[i/3]  (expanded diagonal of the mass Hessian)
__global__ void k_mdiag(const float* __restrict__ m, float* __restrict__ md, int N3) {
  int i = blockIdx.x * TPB + threadIdx.x;
  if (i < N3) md[i] = M_ALPHA * m[i / 3];
}

__global__ void k_init_xi(const float* __restrict__ x, float* __restrict__ xi, int N3) {
  long base = (long)blockIdx.y * N3;
  int i = (blockIdx.x * TPB + threadIdx.x) * 4;
  if (i + 3 < N3) {
    *(float4*)(xi + base + i) = *(const float4*)(x + base + i);
  } else {
    for (int j = i; j < N3; ++j) xi[base + j] = x[base + j];
  }
}

// ---------------- per-Newton kernels ---------------------------------------

// r = md * (xi - ci); dv = 0; zero per-batch scalars
__global__ void k_grad_init(const float* __restrict__ xi, const float* __restrict__ ci,
                            const float* __restrict__ md, float* __restrict__ r,
                            float* __restrict__ dv, float* __restrict__ scal, int N3) {
  int b = blockIdx.y;
  long base = (long)b * N3;
  int i = (blockIdx.x * TPB + threadIdx.x) * 4;
  if (i + 3 < N3) {
    float4 xv = *(const float4*)(xi + base + i);
    float4 cv = *(const float4*)(ci + base + i);
    float4 mv = *(const float4*)(md + i);
    float4 rv;
    rv.x = mv.x * (xv.x - cv.x);
    rv.y = mv.y * (xv.y - cv.y);
    rv.z = mv.z * (xv.z - cv.z);
    rv.w = mv.w * (xv.w - cv.w);
    *(float4*)(r + base + i) = rv;
    *(float4*)(dv + base + i) = make_float4(0.f, 0.f, 0.f, 0.f);
  } else {
    for (int j = i; j < N3; ++j) {
      r[base + j] = md[j] * (xi[base + j] - ci[base + j]);
      dv[base + j] = 0.f;
    }
  }
  if (blockIdx.x == 0 && threadIdx.x < 4) scal[b * 4 + threadIdx.x] = 0.0f;
}

__global__ void k_grad_edges(const float* __restrict__ xi, const int* __restrict__ edges,
                             const float* __restrict__ l0, const float* __restrict__ kk,
                             float* __restrict__ r, int N3, int E) {
  int b = blockIdx.y;
  const float* X = xi + (long)b * N3;
  float* R = r + (long)b * N3;
  int e = blockIdx.x * TPB + threadIdx.x;
  if (e >= E) return;
  int2 pr = ((const int2*)edges)[e];
  int i0 = pr.x, i1 = pr.y;
  float3 a = ld3(X, i0), bb = ld3(X, i1);
  float d0 = a.x - bb.x, d1 = a.y - bb.y, d2 = a.z - bb.z;
  float ln = sqrtf(d0 * d0 + d1 * d1 + d2 * d2 + EPS);
  float c = kk[e] * (ln - l0[e]) / ln;
  atomAdd(&R[3 * i0 + 0], c * d0);
  atomAdd(&R[3 * i0 + 1], c * d1);
  atomAdd(&R[3 * i0 + 2], c * d2);
  atomAdd(&R[3 * i1 + 0], -c * d0);
  atomAdd(&R[3 * i1 + 1], -c * d1);
  atomAdd(&R[3 * i1 + 2], -c * d2);
}

// p = r ; hp = 0 (ready for first HVP) ; rs = <r,r>
__global__ void k_cg_start(const float* __restrict__ r, float* __restrict__ pv,
                           float* __restrict__ hp, float* __restrict__ scal, int N3) {
  int b = blockIdx.y;
  long base = (long)b * N3;
  int i = (blockIdx.x * TPB + threadIdx.x) * 4;
  float v = 0.0f;
  if (i + 3 < N3) {
    float4 rv = *(const float4*)(r + base + i);
    *(float4*)(pv + base + i) = rv;
    *(float4*)(hp + base + i) = make_float4(0.f, 0.f, 0.f, 0.f);
    v = rv.x * rv.x + rv.y * rv.y + rv.z * rv.z + rv.w * rv.w;
  } else {
    for (int j = i; j < N3; ++j) {
      float t = r[base + j];
      pv[base + j] = t;
      hp[base + j] = 0.f;
      v += t * t;
    }
  }
  float s = blockReduceSum(v);
  if (threadIdx.x == 0) atomAdd(&scal[b * 4 + 0], s);
}

// ---------------- CG iteration kernels -------------------------------------

// HVP edge term (adds into zeroed hp). Edge-index stream staged through LDS
// via the CDNA5 async global->LDS path (ASYNCcnt + s_wait_asynccnt).
__global__ void k_hvp_edges(const float* __restrict__ xi, const float* __restrict__ p,
                            const int* __restrict__ edges, const float* __restrict__ l0,
                            const float* __restrict__ kk, float* __restrict__ hp,
                            int N3, int E) {
  int b = blockIdx.y;
  const float* X = xi + (long)b * N3;
  const float* P = p + (long)b * N3;
  float* H = hp + (long)b * N3;
  int e = blockIdx.x * TPB + threadIdx.x;
  int2 pr;
#if HAVE_ASYNC_LDS
  __shared__ int2 sE[TPB];
  {
    int esrc = (e < E) ? e : (E - 1);
    __builtin_amdgcn_global_load_async_to_lds_b64(
        (g_v2i_ptr)(((const int2*)edges) + esrc),
        (l_v2i_ptr)&sE[threadIdx.x],
        0, 0);
    __builtin_amdgcn_s_wait_asynccnt(0);
  }
  __syncthreads();
  pr = sE[threadIdx.x];
#else
  pr = (e < E) ? ((const int2*)edges)[e] : make_int2(0, 0);
#endif
  if (e >= E) return;
  int i0 = pr.x, i1 = pr.y;
  float3 xa = ld3(X, i0), xb = ld3(X, i1);
  float d0 = xa.x - xb.x, d1 = xa.y - xb.y, d2 = xa.z - xb.z;
  float ln = sqrtf(d0 * d0 + d1 * d1 + d2 * d2 + EPS);
  float inv = 1.0f / ln;
  float ke = kk[e], le = l0[e];
  float cA = ke * (1.0f - le * inv);
  float cB = ke * le * inv * inv * inv;
  float3 pa = ld3(P, i0), pb = ld3(P, i1);
  float p0 = pa.x - pb.x, p1 = pa.y - pb.y, p2 = pa.z - pb.z;
  float dd = d0 * p0 + d1 * p1 + d2 * p2;
  float t0 = cA * p0 + cB * d0 * dd;
  float t1 = cA * p1 + cB * d1 * dd;
  float t2 = cA * p2 + cB * d2 * dd;
  atomAdd(&H[3 * i0 + 0], t0);
  atomAdd(&H[3 * i0 + 1], t1);
  atomAdd(&H[3 * i0 + 2], t2);
  atomAdd(&H[3 * i1 + 0], -t0);
  atomAdd(&H[3 * i1 + 1], -t1);
  atomAdd(&H[3 * i1 + 2], -t2);
}

// Axial term is linear: gradient (vin=xi, out=r) and HVP (vin=p, out=hp)
// share one kernel. Topology loaded as b128.
__global__ void k_axial(const float* __restrict__ vin, const int* __restrict__ ai,
                        const float* __restrict__ aw, const float* __restrict__ ak,
                        float* __restrict__ out, int N3, int A) {
  int b = blockIdx.y;
  const float* V = vin + (long)b * N3;
  float* O = out + (long)b * N3;
  int a = blockIdx.x * TPB + threadIdx.x;
  if (a >= A) return;
  int4 id4 = ((const int4*)ai)[a];
  float4 w4 = ((const float4*)aw)[a];
  int idx[4] = {id4.x, id4.y, id4.z, id4.w};
  float w[4] = {w4.x, w4.y, w4.z, w4.w};
  float ax0 = 0.f, ax1 = 0.f, ax2 = 0.f;
#pragma unroll
  for (int q = 0; q < 4; ++q) {
    float3 xv = ld3(V, idx[q]);
    ax0 += w[q] * xv.x;
    ax1 += w[q] * xv.y;
    ax2 += w[q] * xv.z;
  }
  float s = ak[a];
#pragma unroll
  for (int q = 0; q < 4; ++q) {
    float wq = s * w[q];
    atomAdd(&O[3 * idx[q] + 0], wq * ax0);
    atomAdd(&O[3 * idx[q] + 1], wq * ax1);
    atomAdd(&O[3 * idx[q] + 2], wq * ax2);
  }
}

// pHp = <p, hp_sparse + md*p>   (diagonal folded in analytically)
__global__ void k_dot_pHp(const float* __restrict__ p, const float* __restrict__ hp,
                          const float* __restrict__ md, float* __restrict__ scal, int N3) {
  int b = blockIdx.y;
  long base = (long)b * N3;
  int i = (blockIdx.x * TPB + threadIdx.x) * 4;
  float v = 0.0f;
  if (i + 3 < N3) {
    float4 p4 = *(const float4*)(p + base + i);
    float4 h4 = *(const float4*)(hp + base + i);
    float4 m4 = *(const float4*)(md + i);
    v = p4.x * (h4.x + m4.x * p4.x) + p4.y * (h4.y + m4.y * p4.y) +
        p4.z * (h4.z + m4.z * p4.z) + p4.w * (h4.w + m4.w * p4.w);
  } else {
    for (int j = i; j < N3; ++j) {
      float pj = p[base + j];
      v += pj * (hp[base + j] + md[j] * pj);
    }
  }
  float s = blockReduceSum(v);
  if (threadIdx.x == 0) atomAdd(&scal[b * 4 + 1], s);
}

// alpha = rs/(pHp+eps); dv += alpha*p; r -= alpha*(hp_sparse + md*p); rsn = <r,r>
__global__ void k_cg_update(const float* __restrict__ p, const float* __restrict__ hp,
                            const float* __restrict__ md, float* __restrict__ dv,
                            float* __restrict__ r, float* __restrict__ scal, int N3) {
  int b = blockIdx.y;
  long base = (long)b * N3;
  float alpha = scal[b * 4 + 0] / (scal[b * 4 + 1] + EPS);
  int i = (blockIdx.x * TPB + threadIdx.x) * 4;
  float v = 0.0f;
  if (i + 3 < N3) {
    float4 p4 = *(const float4*)(p + base + i);
    float4 h4 = *(const float4*)(hp + base + i);
    float4 m4 = *(const float4*)(md + i);
    float4 d4 = *(float4*)(dv + base + i);
    float4 r4 = *(float4*)(r + base + i);
    d4.x += alpha * p4.x; d4.y += alpha * p4.y;
    d4.z += alpha * p4.z; d4.w += alpha * p4.w;
    r4.x -= alpha * (h4.x + m4.x * p4.x);
    r4.y -= alpha * (h4.y + m4.y * p4.y);
    r4.z -= alpha * (h4.z + m4.z * p4.z);
    r4.w -= alpha * (h4.w + m4.w * p4.w);
    *(float4*)(dv + base + i) = d4;
    *(float4*)(r + base + i) = r4;
    v = r4.x * r4.x + r4.y * r4.y + r4.z * r4.z + r4.w * r4.w;
  } else {
    for (int j = i; j < N3; ++j) {
      float pj = p[base + j];
      dv[base + j] += alpha * pj;
      float rn = r[base + j] - alpha * (hp[base + j] + md[j] * pj);
      r[base + j] = rn;
      v += rn * rn;
    }
  }
  float s = blockReduceSum(v);
  if (threadIdx.x == 0) atomAdd(&scal[b * 4 + 2], s);
}

// p = r + beta*p; hp = 0 (ready for next HVP)
__global__ void k_cg_p(const float* __restrict__ r, float* __restrict__ pv,
                       float* __restrict__ hp, const float* __restrict__ scal, int N3) {
  int b = blockIdx.y;
  long base = (long)b * N3;
  float beta = scal[b * 4 + 2] / (scal[b * 4 + 0] + EPS);
  int i = (blockIdx.x * TPB + threadIdx.x) * 4;
  if (i + 3 < N3) {
    float4 r4 = *(const float4*)(r + base + i);
    float4 p4 = *(float4*)(pv + base + i);
    p4.x = r4.x + beta * p4.x;
    p4.y = r4.y + beta * p4.y;
    p4.z = r4.z + beta * p4.z;
    p4.w = r4.w + beta * p4.w;
    *(float4*)(pv + base + i) = p4;
    *(float4*)(hp + base + i) = make_float4(0.f, 0.f, 0.f, 0.f);
  } else {
    for (int j = i; j < N3; ++j) {
      pv[base + j] = r[base + j] + beta * pv[base + j];
      hp[base + j] = 0.f;
    }
  }
}

__global__ void k_scal_roll(float* __restrict__ scal, int B) {
  int b = threadIdx.x;
  if (b < B) {
    scal[b * 4 + 0] = scal[b * 4 + 2];  // rs <- rs_new
    scal[b * 4 + 1] = 0.0f;             // pHp
    scal[b * 4 + 2] = 0.0f;             // rs_new
  }
}

__global__ void k_x_update(float* __restrict__ xi, const float* __restrict__ dv, int N3) {
  long base = (long)blockIdx.y * N3;
  int i = (blockIdx.x * TPB + threadIdx.x) * 4;
  if (i + 3 < N3) {
    float4 x4 = *(float4*)(xi + base + i);
    float4 d4 = *(const float4*)(dv + base + i);
    x4.x -= d4.x; x4.y -= d4.y; x4.z -= d4.z; x4.w -= d4.w;
    *(float4*)(xi + base + i) = x4;
  } else {
    for (int j = i; j < N3; ++j) xi[base + j] -= dv[base + j];
  }
}

// ---------------------------------------------------------------------------

extern "C" void kernel_launch(void* const* d_in, const int* in_sizes, int n_in,
                              void* d_out, int out_size, void* d_ws, size_t ws_size,
                              hipStream_t stream) {
  const float* x = (const float*)d_in[0];   // (B,N,3) — also serves as ci
  const float* m = (const float*)d_in[1];   // (N,1)
  const int* edges = (const int*)d_in[2];   // (E,2)
  const float* l0 = (const float*)d_in[3];  // (E,1)
  const float* kk = (const float*)d_in[4];  // (E,1)
  const int* ai = (const int*)d_in[5];      // (A,4)
  const float* aw = (const float*)d_in[6];  // (A,4)
  const float* ak = (const float*)d_in[7];  // (A,1)
  float* xi = (float*)d_out;                // working iterate lives in d_out

  const int N = in_sizes[1];
  const int E = in_sizes[3];
  const int A = in_sizes[7];
  const int N3 = 3 * N;
  const int B = in_sizes[0] / N3;
  const size_t BN3 = (size_t)B * N3;

  float* r = (float*)d_ws;       // residual
  float* pv = r + BN3;           // CG direction
  float* hp = pv + BN3;          // H*p (sparse part)
  float* dv = hp + BN3;          // accumulated Newton step d
  float* md = dv + BN3;          // expanded mass diagonal (N3)
  float* scal = md + N3;         // per-batch {rs, pHp, rs_new, pad}

  const int N4 = (N3 + 3) / 4;
  dim3 gV4((unsigned)((N4 + TPB - 1) / TPB), (unsigned)B);
  dim3 gE((unsigned)((E + TPB - 1) / TPB), (unsigned)B);
  dim3 gA((unsigned)((A + TPB - 1) / TPB), (unsigned)B);
  unsigned gS = (unsigned)((N3 + TPB - 1) / TPB);

  k_mdiag<<<gS, TPB, 0, stream>>>(m, md, N3);
  k_init_xi<<<gV4, TPB, 0, stream>>>(x, xi, N3);

  for (int newton = 0; newton < 2; ++newton) {
    // g -> r  (also zeros dv and per-batch scalars)
    k_grad_init<<<gV4, TPB, 0, stream>>>(xi, x, md, r, dv, scal, N3);
    k_grad_edges<<<gE, TPB, 0, stream>>>(xi, edges, l0, kk, r, N3, E);
    k_axial<<<gA, TPB, 0, stream>>>(xi, ai, aw, ak, r, N3, A);
    // p = r ; hp = 0 ; rs = <r,r>
    k_cg_start<<<gV4, TPB, 0, stream>>>(r, pv, hp, scal, N3);

    for (int it = 0; it < 20; ++it) {
      k_hvp_edges<<<gE, TPB, 0, stream>>>(xi, pv, edges, l0, kk, hp, N3, E);
      k_axial<<<gA, TPB, 0, stream>>>(pv, ai, aw, ak, hp, N3, A);
      k_dot_pHp<<<gV4, TPB, 0, stream>>>(pv, hp, md, scal, N3);
      k_cg_update<<<gV4, TPB, 0, stream>>>(pv, hp, md, dv, r, scal, N3);
      k_cg_p<<<gV4, TPB, 0, stream>>>(r, pv, hp, scal, N3);
      k_scal_roll<<<1, 32, 0, stream>>>(scal, B);
    }
    k_x_update<<<gV4, TPB, 0, stream>>>(xi, dv, N3);
  }
}